// Correlation_13675175871187
// MI455X (gfx1250) — compile-verified
//
#include <hip/hip_runtime.h>

typedef float v2f __attribute__((ext_vector_type(2)));
typedef float v8f __attribute__((ext_vector_type(8)));

#define Bn 8
#define Cn 256
#define Hn 64
#define Wn 128
#define HWn (Hn * Wn)

// One wave (= one 32-thread block) computes the full 81-displacement
// correlation for a 4x4 patch of fmap1 pixels via nine f32 WMMA pixel-pair
// GEMMs against the nine 4x4 patches tiling the 12x12 fmap2 halo region.
//
// Interior tiles (82%): all nine patches are in-bounds, and their addresses
// are affine in compile-time constants (ty*4*Wn + tx*4 + c*HWn), so the whole
// body runs off TWO per-lane addresses with immediate offsets — no masking.
// Border tiles: per-patch clamped offsets + v_cndmask zero-padding.
__global__ __launch_bounds__(32) void corr_wmma_f32(
    const float* __restrict__ f1, const float* __restrict__ f2,
    float* __restrict__ out) {
  const int lane = threadIdx.x;        // 0..31
  const int tile = blockIdx.x;         // 0..4095  (uniform)
  const int b    = tile >> 9;          // 16*32 tiles per batch
  const int rem  = tile & 511;
  const int y0   = (rem >> 5) << 2;
  const int x0   = (rem & 31) << 2;

  const int n16  = lane & 15;          // A: M index; B/C: N index
  const int half = lane >> 4;          // K half (channels 2*half, 2*half+1)
  const int py   = n16 >> 2;           // pixel-in-patch row
  const int px   = n16 & 3;            // pixel-in-patch col

  const float* b1 = f1 + (size_t)b * Cn * HWn;   // uniform (SGPR) bases
  const float* b2 = f2 + (size_t)b * Cn * HWn;

  const int off1 = 2 * half * HWn + (y0 + py) * Wn + (x0 + px);

  v8f acc[9];
#pragma unroll
  for (int t = 0; t < 9; ++t)
    acc[t] = (v8f){0.f, 0.f, 0.f, 0.f, 0.f, 0.f, 0.f, 0.f};

  const bool interior =
      (y0 >= 4) && (y0 <= Hn - 8) && (x0 >= 4) && (x0 <= Wn - 8);

  if (interior) {
    // ---- hot path: no boundary handling, all offsets are immediates ----
    const float* pA = b1 + off1;
    const float* pB = b2 + (2 * half * HWn + (y0 - 4 + py) * Wn + (x0 - 4 + px));
#pragma unroll 4
    for (int c0 = 0; c0 < Cn; c0 += 4) {
      v2f a;
      a.x = pA[c0 * HWn];
      a.y = pA[c0 * HWn + HWn];
#pragma unroll
      for (int t = 0; t < 9; ++t) {
        const int ty = t / 3, tx = t % 3;
        const int to = ty * 4 * Wn + tx * 4;   // compile-time constant
        v2f bb;
        bb.x = pB[c0 * HWn + to];
        bb.y = pB[c0 * HWn + to + HWn];
        acc[t] = __builtin_amdgcn_wmma_f32_16x16x4_f32(
            false, a, false, bb, (short)0, acc[t], false, false);
      }
    }
  } else {
    // ---- border path: clamped loads + cndmask zero-padding ----
    int  off2[9];
    bool ib[9];
#pragma unroll
    for (int t = 0; t < 9; ++t) {
      const int ty  = t / 3, tx = t % 3;
      const int p2y = y0 - 4 + ty * 4 + py;
      const int p2x = x0 - 4 + tx * 4 + px;
      ib[t] = ((unsigned)p2y < (unsigned)Hn) && ((unsigned)p2x < (unsigned)Wn);
      const int cy = p2y < 0 ? 0 : (p2y > Hn - 1 ? Hn - 1 : p2y);
      const int cx = p2x < 0 ? 0 : (p2x > Wn - 1 ? Wn - 1 : p2x);
      off2[t] = 2 * half * HWn + cy * Wn + cx;   // always a valid address
    }
#pragma unroll 2
    for (int c0 = 0; c0 < Cn; c0 += 4) {
      v2f a;
      a.x = b1[off1 + c0 * HWn];
      a.y = b1[off1 + c0 * HWn + HWn];
#pragma unroll
      for (int t = 0; t < 9; ++t) {
        const float vx = b2[off2[t] + c0 * HWn];
        const float vy = b2[off2[t] + c0 * HWn + HWn];
        v2f bb;
        bb.x = ib[t] ? vx : 0.0f;   // zero-padding
        bb.y = ib[t] ? vy : 0.0f;
        acc[t] = __builtin_amdgcn_wmma_f32_16x16x4_f32(
            false, a, false, bb, (short)0, acc[t], false, false);
      }
    }
  }

  // Scatter the 81 valid displacement entries out of the 9 C tiles.
  // C layout: row M = r + 8*half, col N = n16 (this lane's f2-patch pixel).
  float* outb = out + (size_t)b * 81 * HWn;    // uniform base, i32 offsets
#pragma unroll
  for (int t = 0; t < 9; ++t) {
    const int ty = t / 3, tx = t % 3;
#pragma unroll
    for (int r = 0; r < 8; ++r) {
      const int i   = r + 8 * half;  // f1 pixel index (row M)
      const int ppy = i >> 2;
      const int ppx = i & 3;
      const unsigned du = (unsigned)(ty * 4 + py - ppy);  // dy + 4
      const unsigned dv = (unsigned)(tx * 4 + px - ppx);  // dx + 4
      if (du <= 8u && dv <= 8u) {
        const int d  = (int)du * 9 + (int)dv;
        const int oo = (d * Hn + (y0 + ppy)) * Wn + (x0 + ppx);
        outb[oo] = acc[t][r];
      }
    }
  }
}

extern "C" void kernel_launch(void* const* d_in, const int* in_sizes, int n_in,
                              void* d_out, int out_size, void* d_ws,
                              size_t ws_size, hipStream_t stream) {
  const float* f1 = (const float*)d_in[0];
  const float* f2 = (const float*)d_in[1];
  float* out = (float*)d_out;
  // 8 batches * 16 * 32 = 4096 tiles; one wave (32 threads) per tile.
  dim3 grid(4096), block(32);
  corr_wmma_f32<<<grid, block, 0, stream>>>(f1, f2, out);
}